// GATClassifier_52621939310632
// MI455X (gfx1250) — compile-verified
//
#include <hip/hip_runtime.h>
#include <hip/hip_bf16.h>

typedef __attribute__((ext_vector_type(2))) float v2f;
typedef __attribute__((ext_vector_type(8))) float v8f;
typedef __attribute__((address_space(3))) float lds_f32;

#define N_NODES  50000
#define N_EDGES  400000
#define N_GRAPHS 64
#define IN_FEATS 256
#define HIDDEN   128
#define HEADS    4
#define NEG_SLOPE 0.2f
#define MAX_K    512

// ---- monotone float <-> u32 ordering (for atomicMax-based segment max) ----
__device__ __forceinline__ unsigned ord_f32(float f) {
  unsigned u = __float_as_uint(f);
  return ((int)u < 0) ? ~u : (u ^ 0x80000000u);
}
__device__ __forceinline__ float unord_f32(unsigned u) {
  u = ((int)u < 0) ? (u ^ 0x80000000u) : ~u;
  return __uint_as_float(u);
}

__global__ void fill_u32(unsigned* __restrict__ p, unsigned v, int n) {
  int i = blockIdx.x * blockDim.x + threadIdx.x;
  if (i < n) p[i] = v;
}

// ---------------------------------------------------------------------------
// C[M,N] = A[M,K] @ B[K,N]  (fp32, V_WMMA_F32_16X16X4_F32)
// blockDim = (32,4): 4 waves, each computes one 16x16 tile; all 4 waves share
// the same N-tile, so the K x 16 strip of B is staged once per block into LDS
// via GLOBAL_LOAD_ASYNC_TO_LDS_B128 (ASYNCcnt), then read back with ds loads.
//
// A 16x4 layout: lanes 0-15 hold (row=m0+lane, K=k0..k0+1),
//                lanes 16-31 hold (row=m0+lane-16, K=k0+2..k0+3)  -> float2 load
// B 4x16 layout: lanes 0-15 hold (col=n0+lane, rows k0,k0+1),
//                lanes 16-31 hold (col=n0+lane-16, rows k0+2,k0+3)
// C/D: VGPR j -> row m0 + j + 8*(lane/16), col n0 + (lane&15)
// ---------------------------------------------------------------------------
__global__ void __launch_bounds__(128)
gemm_wmma_f32(const float* __restrict__ A, const float* __restrict__ B,
              float* __restrict__ C, int M, int N, int K) {
  __shared__ float Blds[MAX_K * 16];

  const int tid  = threadIdx.y * 32 + threadIdx.x;
  const int lane = threadIdx.x & 31;
  const int hi   = lane >> 4;     // half-wave select
  const int lo   = lane & 15;
  const int n0   = blockIdx.x * 16;

  // ---- async-stage B[0..K-1, n0..n0+15] into LDS (16B per lane per issue) ----
  {
    const int quarter = tid & 3;       // which 4-float chunk of the 16-wide row
    const int krow0   = tid >> 2;      // starting row; 32 rows covered per pass
    for (int k = krow0; k < K; k += 32) {
      unsigned long long gaddr =
          (unsigned long long)(const void*)(B + (size_t)k * N + n0 + quarter * 4);
      unsigned laddr = (unsigned)(unsigned long long)(lds_f32*)&Blds[k * 16 + quarter * 4];
      asm volatile("global_load_async_to_lds_b128 %0, %1, off"
                   :: "v"(laddr), "v"(gaddr) : "memory");
    }
    asm volatile("s_wait_asynccnt 0x0" ::: "memory");
  }
  __syncthreads();

  const int tileM = blockIdx.y * 4 + threadIdx.y;
  const int m0    = tileM * 16;
  if (m0 >= M) return;

  v8f acc = {};
  const float* Arow = A + (size_t)(m0 + lo) * K + 2 * hi;
  const float* Bl   = &Blds[(2 * hi) * 16 + lo];

  #pragma unroll 4
  for (int k0 = 0; k0 < K; k0 += 4) {
    v2f a, b;
    a.x = Arow[k0];
    a.y = Arow[k0 + 1];
    b.x = Bl[k0 * 16];        // row k0+2*hi   of the strip
    b.y = Bl[k0 * 16 + 16];   // row k0+2*hi+1 of the strip
    acc = __builtin_amdgcn_wmma_f32_16x16x4_f32(
        /*neg_a=*/false, a, /*neg_b=*/false, b,
        /*c_mod=*/(short)0, acc, /*reuse_a=*/false, /*reuse_b=*/false);
  }

  float* Cp = C + (size_t)(m0 + 8 * hi) * N + n0 + lo;
  #pragma unroll
  for (int j = 0; j < 8; ++j) Cp[(size_t)j * N] = acc[j];
}

// el[n,h] = sum_d feat[n,h,d]*al[h,d] ; er likewise. One wave per (n,h).
__global__ void attn_scores(const float* __restrict__ feat,
                            const float* __restrict__ al, const float* __restrict__ ar,
                            float* __restrict__ el, float* __restrict__ er,
                            int N, int H, int D) {
  int unit = (blockIdx.x * blockDim.x + threadIdx.x) >> 5;
  int lane = threadIdx.x & 31;
  if (unit >= N * H) return;
  int n = unit / H, h = unit - n * H;
  const float* f   = feat + ((size_t)n * H + h) * D;
  const float* alp = al + (size_t)h * D;
  const float* arp = ar + (size_t)h * D;
  float sl = 0.f, sr = 0.f;
  for (int d = lane; d < D; d += 32) {
    float v = f[d];
    sl += v * alp[d];
    sr += v * arp[d];
  }
  #pragma unroll
  for (int off = 16; off > 0; off >>= 1) {
    sl += __shfl_down(sl, off, 32);
    sr += __shfl_down(sr, off, 32);
  }
  if (lane == 0) { el[unit] = sl; er[unit] = sr; }
}

// e = leaky_relu(el[src]+er[dst]); segment max via ordered-u32 atomicMax
__global__ void edge_pass1(const int* __restrict__ src, const int* __restrict__ dst,
                           const float* __restrict__ el, const float* __restrict__ er,
                           float* __restrict__ ebuf, unsigned* __restrict__ mbits,
                           int E, int H) {
  int t = blockIdx.x * blockDim.x + threadIdx.x;
  if (t >= E * H) return;
  int e = t / H, h = t - e * H;
  int s = src[e], d = dst[e];
  float x = el[s * H + h] + er[d * H + h];
  x = (x > 0.f) ? x : NEG_SLOPE * x;
  ebuf[t] = x;
  atomicMax(&mbits[d * H + h], ord_f32(x));
}

// ex = exp(e - m[dst]); denom[dst] += ex
__global__ void edge_pass2(const int* __restrict__ dst, float* __restrict__ ebuf,
                           const unsigned* __restrict__ mbits, float* __restrict__ denom,
                           int E, int H) {
  int t = blockIdx.x * blockDim.x + threadIdx.x;
  if (t >= E * H) return;
  int e = t / H, h = t - e * H;
  int d = dst[e];
  float ex = expf(ebuf[t] - unord_f32(mbits[d * H + h]));
  ebuf[t] = ex;
  atomicAdd(&denom[d * H + h], ex);
}

// agg[dst,h,:] += (ex/denom[dst,h]) * feat[src,h,:] ; one wave per (edge,head)
__global__ void edge_pass3(const int* __restrict__ src, const int* __restrict__ dst,
                           const float* __restrict__ ebuf, const float* __restrict__ denom,
                           const float* __restrict__ feat, float* __restrict__ agg,
                           int E, int H, int D) {
  int unit = (blockIdx.x * blockDim.x + threadIdx.x) >> 5;
  int lane = threadIdx.x & 31;
  if (unit >= E * H) return;
  int e = unit / H, h = unit - e * H;
  int s = src[e], d = dst[e];
  float a = ebuf[unit] / denom[d * H + h];
  const float* fs = feat + ((size_t)s * H + h) * D;
  float* ag       = agg  + ((size_t)d * H + h) * D;
  for (int c = lane; c < D; c += 32) atomicAdd(&ag[c], a * fs[c]);
}

// h = (agg + b) with optional ELU, in place
__global__ void bias_act(float* __restrict__ h, const float* __restrict__ b,
                         int total, int C, int do_elu) {
  int i = blockIdx.x * blockDim.x + threadIdx.x;
  if (i >= total) return;
  float v = h[i] + b[i % C];
  if (do_elu) v = (v > 0.f) ? v : (expf(v) - 1.0f);
  h[i] = v;
}

// per-graph sum + count
__global__ void pool_accum(const float* __restrict__ h, const int* __restrict__ gid,
                           float* __restrict__ sums, float* __restrict__ cnts,
                           int N, int D) {
  int i = blockIdx.x * blockDim.x + threadIdx.x;
  if (i >= N * D) return;
  int n = i / D, d = i - n * D;
  int g = gid[n];
  atomicAdd(&sums[g * D + d], h[i]);
  if (d == 0) atomicAdd(&cnts[g], 1.0f);
}

// y[g] = sigmoid(mean_g . fc_w + fc_b) ; one wave per graph
__global__ void final_fc(const float* __restrict__ sums, const float* __restrict__ cnts,
                         const float* __restrict__ fc_w, const float* __restrict__ fc_b,
                         float* __restrict__ out, int D) {
  int g = blockIdx.x;
  int lane = threadIdx.x;
  float c = cnts[g];
  c = (c > 1.f) ? c : 1.f;
  float s = 0.f;
  for (int d = lane; d < D; d += 32) s += (sums[g * D + d] / c) * fc_w[d];
  #pragma unroll
  for (int off = 16; off > 0; off >>= 1) s += __shfl_down(s, off, 32);
  if (lane == 0) {
    float z = s + fc_b[0];
    out[g] = 1.0f / (1.0f + expf(-z));
  }
}

extern "C" void kernel_launch(void* const* d_in, const int* in_sizes, int n_in,
                              void* d_out, int out_size, void* d_ws, size_t ws_size,
                              hipStream_t stream) {
  const float* features = (const float*)d_in[0];
  const int*   src      = (const int*)d_in[1];
  const int*   dst      = (const int*)d_in[2];
  const int*   gid      = (const int*)d_in[3];
  const float* W1   = (const float*)d_in[4];
  const float* al1  = (const float*)d_in[5];
  const float* ar1  = (const float*)d_in[6];
  const float* b1   = (const float*)d_in[7];
  const float* W2   = (const float*)d_in[8];
  const float* al2  = (const float*)d_in[9];
  const float* ar2  = (const float*)d_in[10];
  const float* b2   = (const float*)d_in[11];
  const float* fc_w = (const float*)d_in[12];
  const float* fc_b = (const float*)d_in[13];
  float* out = (float*)d_out;
  (void)in_sizes; (void)n_in; (void)out_size; (void)ws_size;

  // ---- workspace layout ----
  char* ws = (char*)d_ws;
  size_t off = 0;
  auto alloc = [&](size_t bytes) -> void* {
    void* p = ws + off;
    off += (bytes + 255) & ~(size_t)255;
    return p;
  };
  const int F1 = HEADS * HIDDEN;  // 512
  float*    feat1  = (float*)alloc((size_t)N_NODES * F1 * 4);
  float*    h1     = (float*)alloc((size_t)N_NODES * F1 * 4);     // agg1 -> elu in place
  float*    feat2  = (float*)alloc((size_t)N_NODES * HIDDEN * 4);
  float*    agg2   = (float*)alloc((size_t)N_NODES * HIDDEN * 4);
  float*    el1    = (float*)alloc((size_t)N_NODES * HEADS * 4);
  float*    er1    = (float*)alloc((size_t)N_NODES * HEADS * 4);
  unsigned* m1     = (unsigned*)alloc((size_t)N_NODES * HEADS * 4);
  float*    den1   = (float*)alloc((size_t)N_NODES * HEADS * 4);
  float*    el2    = (float*)alloc((size_t)N_NODES * 4);
  float*    er2    = (float*)alloc((size_t)N_NODES * 4);
  unsigned* m2     = (unsigned*)alloc((size_t)N_NODES * 4);
  float*    den2   = (float*)alloc((size_t)N_NODES * 4);
  float*    ebuf   = (float*)alloc((size_t)N_EDGES * HEADS * 4);  // reused for layer 2
  float*    sums   = (float*)alloc((size_t)N_GRAPHS * HIDDEN * 4);
  float*    cnts   = (float*)alloc((size_t)N_GRAPHS * 4);

  auto cdiv = [](long long a, long long b) { return (int)((a + b - 1) / b); };
  #define FILL(ptr, words) fill_u32<<<cdiv((words), 256), 256, 0, stream>>>((unsigned*)(ptr), 0u, (int)(words))

  FILL(h1,   (size_t)N_NODES * F1);
  FILL(agg2, (size_t)N_NODES * HIDDEN);
  FILL(m1,   (size_t)N_NODES * HEADS);
  FILL(den1, (size_t)N_NODES * HEADS);
  FILL(m2,   (size_t)N_NODES);
  FILL(den2, (size_t)N_NODES);
  FILL(sums, (size_t)N_GRAPHS * HIDDEN);
  FILL(cnts, (size_t)N_GRAPHS);

  const int mtiles = cdiv(N_NODES, 16);          // 3125
  dim3 gblk(32, 4);

  // ---- layer 1 ----
  gemm_wmma_f32<<<dim3(F1 / 16, cdiv(mtiles, 4)), gblk, 0, stream>>>(
      features, W1, feat1, N_NODES, F1, IN_FEATS);
  attn_scores<<<cdiv((long long)N_NODES * HEADS * 32, 256), 256, 0, stream>>>(
      feat1, al1, ar1, el1, er1, N_NODES, HEADS, HIDDEN);
  edge_pass1<<<cdiv((long long)N_EDGES * HEADS, 256), 256, 0, stream>>>(
      src, dst, el1, er1, ebuf, m1, N_EDGES, HEADS);
  edge_pass2<<<cdiv((long long)N_EDGES * HEADS, 256), 256, 0, stream>>>(
      dst, ebuf, m1, den1, N_EDGES, HEADS);
  edge_pass3<<<cdiv((long long)N_EDGES * HEADS * 32, 256), 256, 0, stream>>>(
      src, dst, ebuf, den1, feat1, h1, N_EDGES, HEADS, HIDDEN);
  bias_act<<<cdiv((long long)N_NODES * F1, 256), 256, 0, stream>>>(
      h1, b1, N_NODES * F1, F1, /*elu=*/1);

  // ---- layer 2 (1 head) ----
  gemm_wmma_f32<<<dim3(HIDDEN / 16, cdiv(mtiles, 4)), gblk, 0, stream>>>(
      h1, W2, feat2, N_NODES, HIDDEN, F1);
  attn_scores<<<cdiv((long long)N_NODES * 32, 256), 256, 0, stream>>>(
      feat2, al2, ar2, el2, er2, N_NODES, 1, HIDDEN);
  edge_pass1<<<cdiv((long long)N_EDGES, 256), 256, 0, stream>>>(
      src, dst, el2, er2, ebuf, m2, N_EDGES, 1);
  edge_pass2<<<cdiv((long long)N_EDGES, 256), 256, 0, stream>>>(
      dst, ebuf, m2, den2, N_EDGES, 1);
  edge_pass3<<<cdiv((long long)N_EDGES * 32, 256), 256, 0, stream>>>(
      src, dst, ebuf, den2, feat2, agg2, N_EDGES, 1, HIDDEN);
  bias_act<<<cdiv((long long)N_NODES * HIDDEN, 256), 256, 0, stream>>>(
      agg2, b2, N_NODES * HIDDEN, HIDDEN, /*elu=*/0);

  // ---- readout ----
  pool_accum<<<cdiv((long long)N_NODES * HIDDEN, 256), 256, 0, stream>>>(
      agg2, gid, sums, cnts, N_NODES, HIDDEN);
  final_fc<<<N_GRAPHS, 32, 0, stream>>>(sums, cnts, fc_w, fc_b, out, HIDDEN);
  #undef FILL
}